// ESN_3839700763091
// MI455X (gfx1250) — compile-verified
//
#include <hip/hip_runtime.h>
#include <math.h>
#include <stdint.h>

// ---------------------------------------------------------------------------
// Echo State Network recurrence on MI455X (gfx1250), persistent-kernel design.
//
//   x_t = tanh(w_in * u_t + W @ x_{t-1}),  y_t = x_t . w_out   (t >= 200)
//
//  * 125 workgroups x 128 threads (4 wave32 waves). Block b owns rows
//    [16b, 16b+16) of W; the 16x2000 f32 strip (128 KB) is staged into LDS
//    ONCE and reused for all 20000 steps.
//  * Each step, the 8 KB state vector is pulled into LDS with gfx1250
//    GLOBAL_LOAD_ASYNC_TO_LDS_B128 (ASYNCcnt), overlapping the readout phase;
//    the WMMA loop then feeds both operands from LDS.
//  * Mat-vec via V_WMMA_F32_16X16X4_F32: A = 16x4 strip tile (lane=row,
//    VGPR0/1 = K pair per ISA layout), B = x[k..k+3] replicated across the 16
//    columns -> every D column equals the 16-row mat-vec partial. 4 waves
//    split K=2000 into 500-chunks (125 WMMAs each), reduced in LDS.
//  * One device-wide epoch barrier per step (all global stores issue from
//    wave 0, which also runs the barrier, so the ACQ_REL RMW covers them).
//  * Readout: per-block partial dot via fixed shuffle tree into a
//    double-buffered ws array; block 0 reduces the 125 partials for step t-1
//    during step t (fixed order -> bitwise deterministic, no float atomics).
// ---------------------------------------------------------------------------

typedef float v2f __attribute__((ext_vector_type(2)));
typedef float v8f __attribute__((ext_vector_type(8)));

#define H          2000
#define T_STEPS    20000
#define WASHOUT    200
#define NBLK       125           // 125 * 16 = 2000 rows
#define NTHREADS   128           // 4 waves of 32
#define NWAVES     4
#define KCHUNK     500           // K columns per wave (500/4 = 125 WMMAs)
#define LDS_STRIDE 2002          // padded row stride (dwords): 2002 % 64 = 18,
                                 // gcd(18,64)=2 -> 16 strip rows hit 16 banks
#define LDS_PART_OFS (16 * LDS_STRIDE)        // 64 floats of partial exchange
#define LDS_X_OFS    (LDS_PART_OFS + 64)      // 2000-float state mirror (16B aligned)
#define LDS_FLOATS   (LDS_X_OFS + 2048)

// workspace layout (bytes)
#define WS_CNT    0
#define WS_EPOCH  4
#define WS_X0     256                     // x double buffer: 2 x 2048 floats
#define WS_PART   (256 + 2 * 2048 * 4)    // partial double buffer: 2 x 128 floats
#define WS_USED   (WS_PART + 2 * 128 * 4)

__device__ __forceinline__ void grid_barrier(unsigned* cnt, unsigned* epoch,
                                             unsigned nblk) {
  __syncthreads();
  if (threadIdx.x == 0) {
    unsigned e = __hip_atomic_load(epoch, __ATOMIC_RELAXED, __HIP_MEMORY_SCOPE_AGENT);
    unsigned arrived =
        __hip_atomic_fetch_add(cnt, 1u, __ATOMIC_ACQ_REL, __HIP_MEMORY_SCOPE_AGENT);
    if (arrived == nblk - 1u) {
      __hip_atomic_store(cnt, 0u, __ATOMIC_RELAXED, __HIP_MEMORY_SCOPE_AGENT);
      __hip_atomic_store(epoch, e + 1u, __ATOMIC_RELEASE, __HIP_MEMORY_SCOPE_AGENT);
    } else {
      while (__hip_atomic_load(epoch, __ATOMIC_ACQUIRE, __HIP_MEMORY_SCOPE_AGENT) == e)
        __builtin_amdgcn_s_sleep(1);
    }
  }
  __syncthreads();
}

__global__ __launch_bounds__(NTHREADS, 1) void esn_persistent_kernel(
    const float* __restrict__ u,     const float* __restrict__ w_in,
    const float* __restrict__ w_res, const float* __restrict__ w_out,
    float* __restrict__ out,         unsigned char* __restrict__ ws) {
  extern __shared__ float smem[];
  float* lds_w    = smem;                   // 16 * LDS_STRIDE floats (W strip)
  float* lds_part = smem + LDS_PART_OFS;    // NWAVES * 16 floats
  float* lds_x    = smem + LDS_X_OFS;       // 2000 floats (state mirror)
  // LDS byte offset of lds_x: flat->LDS aperture keeps the offset in addr[31:0]
  const uint32_t lds_x_byteofs = (uint32_t)(uintptr_t)lds_x;

  unsigned* cnt   = (unsigned*)(ws + WS_CNT);
  unsigned* epoch = (unsigned*)(ws + WS_EPOCH);
  float* xbuf0 = (float*)(ws + WS_X0);
  float* xbuf1 = xbuf0 + 2048;
  float* part0 = (float*)(ws + WS_PART);
  float* part1 = part0 + 128;

  const int blk  = blockIdx.x;
  const int tid  = threadIdx.x;
  const int wv   = tid >> 5;           // wave id (wave32!)
  const int lane = tid & 31;
  const int half = lane >> 4;          // 0: lanes 0-15 (K+0,K+1), 1: lanes 16-31 (K+2,K+3)
  const int rloc = lane & 15;          // A-matrix row within strip = lane % 16
  const int row0 = blk * 16;

  // ---- stage this block's 16-row W strip into LDS (reused 20000x) ----------
  for (int r = 0; r < 16; ++r) {
    const float2* src = (const float2*)(w_res + (size_t)(row0 + r) * H);
    float2*       dst = (float2*)(lds_w + r * LDS_STRIDE);
    for (int k2 = tid; k2 < H / 2; k2 += NTHREADS) dst[k2] = src[k2];
  }
  __syncthreads();

  const float* wrow = lds_w + rloc * LDS_STRIDE;   // per-lane row base (LDS)
  const int    kofs = half << 1;                   // per-lane K offset in the 4-tile

  for (int t = 0; t < T_STEPS; ++t) {
    const float* xprev = (t & 1) ? xbuf1 : xbuf0;  // buf0 memset-zero => x_{-1}=0
    float*       xcur  = (t & 1) ? xbuf0 : xbuf1;
    float*       partW = (t & 1) ? part1 : part0;  // partials for step t
    const float* partR = (t & 1) ? part0 : part1;  // partials written at step t-1

    // ---- async-copy x_{t-1} (8 KB) global -> LDS; overlaps readout below ---
    for (int i4 = tid; i4 < H / 4; i4 += NTHREADS) {
      uint32_t ldsoff = lds_x_byteofs + ((uint32_t)i4 << 4);
      uint64_t gaddr  = (uint64_t)(uintptr_t)(xprev + (i4 << 2));
      asm volatile("global_load_async_to_lds_b128 %0, %1, off"
                   :: "v"(ldsoff), "v"(gaddr) : "memory");
    }

    // ---- block 0: finish readout of step t-1 (fixed order => deterministic)
    if (blk == 0 && wv == 0 && t >= 1) {
      const int s = t - 1;
      float acc = partR[lane] + partR[lane + 32] + partR[lane + 64] + partR[lane + 96];
      for (int m = 16; m >= 1; m >>= 1) acc += __shfl_xor(acc, m, 32);
      if (lane == 0 && s >= WASHOUT) out[s - WASHOUT] = acc;
    }

    // wait for our wave's async copies, then make LDS visible block-wide
    asm volatile("s_wait_asynccnt 0x0" ::: "memory");
    __syncthreads();

    // ---- WMMA mat-vec partial over this wave's K chunk ----------------------
    v8f acc = {0.f, 0.f, 0.f, 0.f, 0.f, 0.f, 0.f, 0.f};
    const int kend = wv * KCHUNK + KCHUNK;
    for (int k = wv * KCHUNK; k < kend; k += 4) {
      const int kk = k + kofs;
      v2f a = *(const v2f*)(wrow + kk);     // ds_load_b64 from W strip
      v2f b = *(const v2f*)(lds_x + kk);    // ds_load_b64 broadcast of state
      acc = __builtin_amdgcn_wmma_f32_16x16x4_f32(
          /*neg_a=*/false, a, /*neg_b=*/false, b,
          /*c_mod=*/(short)0, acc, /*reuse_a=*/false, /*reuse_b=*/false);
    }

    // ---- extract 16 row sums (C/D layout: lane 0 VGPR v = row v, lane 16 = v+8)
    if (lane == 0 || lane == 16) {
      float* dst = lds_part + wv * 16 + half * 8;
#pragma unroll
      for (int i = 0; i < 8; ++i) dst[i] = acc[i];
    }
    __syncthreads();

    // ---- wave 0: cross-wave reduce, tanh, publish x_t and readout partial ---
    if (wv == 0) {
      float pr = 0.f;
      if (lane < 16) {
        float y = lds_part[lane] + lds_part[16 + lane] +
                  lds_part[32 + lane] + lds_part[48 + lane];
        const int r = row0 + lane;
        const float ut = u[t];                       // uniform -> scalar load
        const float xv = tanhf(fmaf(w_in[r], ut, y));
        xcur[r] = xv;
        pr = xv * w_out[r];
      }
      for (int m = 16; m >= 1; m >>= 1) pr += __shfl_xor(pr, m, 32);
      if (lane == 0) partW[blk] = pr;                // slots 125..127 stay 0 (memset)
    }

    grid_barrier(cnt, epoch, NBLK);   // entry __syncthreads guards lds_part/lds_x
  }

  // ---- epilogue: readout for the final step ---------------------------------
  if (blk == 0 && wv == 0) {
    const float* partR = ((T_STEPS - 1) & 1) ? part1 : part0;
    float acc = partR[lane] + partR[lane + 32] + partR[lane + 64] + partR[lane + 96];
    for (int m = 16; m >= 1; m >>= 1) acc += __shfl_xor(acc, m, 32);
    if (lane == 0) out[(T_STEPS - 1) - WASHOUT] = acc;
  }
}

extern "C" void kernel_launch(void* const* d_in, const int* in_sizes, int n_in,
                              void* d_out, int out_size, void* d_ws, size_t ws_size,
                              hipStream_t stream) {
  const float* u     = (const float*)d_in[0];
  const float* w_in  = (const float*)d_in[1];
  const float* w_res = (const float*)d_in[2];
  const float* w_out = (const float*)d_in[3];
  float*       out   = (float*)d_out;

  // Zero barrier state, x double-buffer (=> x_0 starts from zeros) and
  // readout partials. Async memset is graph-capture legal, runs per replay.
  size_t zb = (ws_size < (size_t)WS_USED) ? ws_size : (size_t)WS_USED;
  hipMemsetAsync(d_ws, 0, zb, stream);

  const size_t shmem = (size_t)LDS_FLOATS * sizeof(float);
  hipLaunchKernelGGL(esn_persistent_kernel, dim3(NBLK), dim3(NTHREADS), shmem,
                     stream, u, w_in, w_res, w_out, out,
                     (unsigned char*)d_ws);
}